// GCN_44925357916599
// MI455X (gfx1250) — compile-verified
//
#include <hip/hip_runtime.h>

typedef float v2f __attribute__((ext_vector_type(2)));
typedef float v8f __attribute__((ext_vector_type(8)));

// ---------------------------------------------------------------------------
// Degree / normalization kernels
// ---------------------------------------------------------------------------
__global__ void k_deg_init(float* __restrict__ deg, int n) {
    int i = blockIdx.x * blockDim.x + threadIdx.x;
    if (i < n) deg[i] = 1.0f;  // self-loop contributes 1
}

__global__ void k_deg_count(const int* __restrict__ dst, float* __restrict__ deg, int e) {
    int i = blockIdx.x * blockDim.x + threadIdx.x;
    if (i < e) atomicAdd(&deg[dst[i]], 1.0f);
}

__global__ void k_dinv(float* __restrict__ deg, int n) {
    int i = blockIdx.x * blockDim.x + threadIdx.x;
    if (i < n) deg[i] = 1.0f / sqrtf(deg[i]);  // deg >= 1 always (self-loop)
}

// ---------------------------------------------------------------------------
// f32 WMMA GEMM: H[nrows, NOUT] = A[nrows, K] * W[K, NOUT]
// One wave computes a 16-row strip across all NOUT columns using
// V_WMMA_F32_16X16X4_F32 (16x16 f32 accumulation, K-step 4).
//
// ISA 7.12.2 32-bit layouts (wave32):
//   A 16x4:  lane L(0..15):  v0=A[L][k+0], v1=A[L][k+1]
//            lane 16+L:      v0=A[L][k+2], v1=A[L][k+3]
//   B 4x16:  lane L(0..15):  v0=B[k+0][n], v1=B[k+1][n]
//            lane 16+L:      v0=B[k+2][n], v1=B[k+3][n]
//   D 16x16: vgpr j, lanes 0-15 -> row j; lanes 16-31 -> row j+8
// ---------------------------------------------------------------------------
template <int K, int NOUT>
__global__ void k_gemm_wmma(const float* __restrict__ A, const float* __restrict__ W,
                            float* __restrict__ H, int nrows) {
    const int wave = blockIdx.x * (blockDim.x >> 5) + (threadIdx.x >> 5);
    const int lane = threadIdx.x & 31;
    const int m    = lane & 15;   // row within 16-half / col within tile
    const int kh   = lane >> 4;   // 0: K pair {0,1}, 1: K pair {2,3}
    const int r0   = wave * 16;
    if (r0 + 16 > nrows) return;  // full strips only (tail handled separately)

    constexpr int NT = NOUT / 16;
    v8f acc[NT];
#pragma unroll
    for (int t = 0; t < NT; ++t)
#pragma unroll
        for (int j = 0; j < 8; ++j) acc[t][j] = 0.0f;

    const float* arow = A + (size_t)(r0 + m) * K;
    for (int k = 0; k < K; k += 4) {
        v2f a;
        a.x = arow[k + 2 * kh + 0];
        a.y = arow[k + 2 * kh + 1];
#pragma unroll
        for (int t = 0; t < NT; ++t) {
            v2f b;
            b.x = W[(size_t)(k + 2 * kh + 0) * NOUT + t * 16 + m];
            b.y = W[(size_t)(k + 2 * kh + 1) * NOUT + t * 16 + m];
            acc[t] = __builtin_amdgcn_wmma_f32_16x16x4_f32(
                /*neg_a=*/false, a, /*neg_b=*/false, b,
                /*c_mod=*/(short)0, acc[t], /*reuse_a=*/false, /*reuse_b=*/false);
        }
    }

#pragma unroll
    for (int t = 0; t < NT; ++t) {
#pragma unroll
        for (int j = 0; j < 8; ++j)
            H[(size_t)(r0 + j + 8 * kh) * NOUT + t * 16 + m] = acc[t][j];
    }
}

// Scalar tail for nrows % 16 != 0 (not hit for N=100000, kept for robustness).
template <int K, int NOUT>
__global__ void k_gemm_tail(const float* __restrict__ A, const float* __restrict__ W,
                            float* __restrict__ H, int row0, int nrows) {
    int idx = blockIdx.x * blockDim.x + threadIdx.x;
    int r = row0 + idx / NOUT;
    int c = idx % NOUT;
    if (r >= nrows) return;
    float s = 0.0f;
    for (int k = 0; k < K; ++k) s += A[(size_t)r * K + k] * W[(size_t)k * NOUT + c];
    H[(size_t)r * NOUT + c] = s;
}

// ---------------------------------------------------------------------------
// agg[i,:] = h[i,:] * dinv[i]^2  (+ bias)   -- self-loop term, initializes agg
// ---------------------------------------------------------------------------
template <int C, bool HAS_BIAS>
__global__ void k_self_init(const float* __restrict__ h, const float* __restrict__ dinv,
                            const float* __restrict__ bias, float* __restrict__ agg, int n) {
    long long idx = (long long)blockIdx.x * blockDim.x + threadIdx.x;
    long long total = (long long)n * C;
    if (idx >= total) return;
    int i = (int)(idx / C);
    int c = (int)(idx % C);
    float d = dinv[i];
    float v = h[idx] * d * d;
    if (HAS_BIAS) v += bias[c];
    agg[idx] = v;
}

// ---------------------------------------------------------------------------
// Edge scatter-add: agg[dst,:] += h[src,:] * dinv[src]*dinv[dst]
// C/4 threads per edge, float4 gather, 4x global_atomic_add_f32 (hits L2;
// the 25.6MB target buffer fits in MI455X's 192MB L2).
// ---------------------------------------------------------------------------
template <int C>
__global__ void k_edge_agg(const int* __restrict__ src, const int* __restrict__ dst,
                           const float* __restrict__ h, const float* __restrict__ dinv,
                           float* __restrict__ agg, int e) {
    constexpr int TPE = C / 4;
    long long tid = (long long)blockIdx.x * blockDim.x + threadIdx.x;
    long long total = (long long)e * TPE;
    if (tid >= total) return;
    int ed = (int)(tid / TPE);
    int q  = (int)(tid % TPE);
    int s = src[ed];
    int d = dst[ed];
    float norm = dinv[s] * dinv[d];
    const float4 v = *((const float4*)(h + (size_t)s * C) + q);
    float* o = agg + (size_t)d * C + q * 4;
    atomicAdd(o + 0, v.x * norm);
    atomicAdd(o + 1, v.y * norm);
    atomicAdd(o + 2, v.z * norm);
    atomicAdd(o + 3, v.w * norm);
}

// ---------------------------------------------------------------------------
// p[i,c] = relu(agg[i,c] + b[c])   (C = 64)
// ---------------------------------------------------------------------------
template <int C>
__global__ void k_relu_bias(const float* __restrict__ agg, const float* __restrict__ b,
                            float* __restrict__ p, int n) {
    long long idx = (long long)blockIdx.x * blockDim.x + threadIdx.x;
    long long total = (long long)n * C;
    if (idx >= total) return;
    int c = (int)(idx % C);
    float v = agg[idx] + b[c];
    p[idx] = v > 0.0f ? v : 0.0f;
}

// ---------------------------------------------------------------------------
// Host-side orchestration
// ---------------------------------------------------------------------------
extern "C" void kernel_launch(void* const* d_in, const int* in_sizes, int n_in,
                              void* d_out, int out_size, void* d_ws, size_t ws_size,
                              hipStream_t stream) {
    const float* x  = (const float*)d_in[0];
    const int*   ei = (const int*)d_in[1];
    const float* W1 = (const float*)d_in[2];
    const float* b1 = (const float*)d_in[3];
    const float* W2 = (const float*)d_in[4];
    const float* b2 = (const float*)d_in[5];
    float* out = (float*)d_out;

    const int n = in_sizes[0] / 128;  // 100000
    const int e = in_sizes[1] / 2;    // 1600000
    const int* src = ei;
    const int* dst = ei + e;

    // Workspace layout (reused across phases):
    //   dinv: n floats | bufA: n*64 floats (h1, then p1) | bufB: n*64 floats (agg1, then h2)
    char* ws = (char*)d_ws;
    size_t off0 = ((size_t)n * sizeof(float) + 255) & ~(size_t)255;
    float* dinv = (float*)ws;
    float* bufA = (float*)(ws + off0);
    float* bufB = bufA + (size_t)n * 64;

    const int tb = 256;
    auto cdiv = [](long long a, long long b) { return (int)((a + b - 1) / b); };

    // --- normalization ---
    k_deg_init<<<cdiv(n, tb), tb, 0, stream>>>(dinv, n);
    k_deg_count<<<cdiv(e, tb), tb, 0, stream>>>(dst, dinv, e);
    k_dinv<<<cdiv(n, tb), tb, 0, stream>>>(dinv, n);

    // --- layer 1: h1 = x @ W1 (WMMA) ---
    const int strips = n / 16;
    const int wavesPerBlk = tb / 32;
    k_gemm_wmma<128, 64><<<cdiv(strips, wavesPerBlk), tb, 0, stream>>>(x, W1, bufA, n);
    if (n % 16) {
        int tail = (n % 16) * 64;
        k_gemm_tail<128, 64><<<cdiv(tail, tb), tb, 0, stream>>>(x, W1, bufA, strips * 16, n);
    }

    // --- layer 1 aggregation ---
    k_self_init<64, false><<<cdiv((long long)n * 64, tb), tb, 0, stream>>>(bufA, dinv, nullptr, bufB, n);
    k_edge_agg<64><<<cdiv((long long)e * 16, tb), tb, 0, stream>>>(src, dst, bufA, dinv, bufB, e);
    k_relu_bias<64><<<cdiv((long long)n * 64, tb), tb, 0, stream>>>(bufB, b1, bufA, n);  // p1 -> bufA

    // --- layer 2: h2 = p1 @ W2 (WMMA) ---
    k_gemm_wmma<64, 32><<<cdiv(strips, wavesPerBlk), tb, 0, stream>>>(bufA, W2, bufB, n);
    if (n % 16) {
        int tail = (n % 16) * 32;
        k_gemm_tail<64, 32><<<cdiv(tail, tb), tb, 0, stream>>>(bufA, W2, bufB, strips * 16, n);
    }

    // --- layer 2 aggregation (bias folded into self-loop init; fully overwrites d_out) ---
    k_self_init<32, true><<<cdiv((long long)n * 32, tb), tb, 0, stream>>>(bufB, dinv, b2, out, n);
    k_edge_agg<32><<<cdiv((long long)e * 8, tb), tb, 0, stream>>>(src, dst, bufB, dinv, out, e);

    (void)n_in; (void)out_size; (void)ws_size;
}